// RPN_24472723653113
// MI455X (gfx1250) — compile-verified
//
#include <hip/hip_runtime.h>
#include <cstdint>
#include <cstddef>

// ---- problem geometry (fixed by the reference) ----
#define NANCH   25200          // total anchors (19200 p3 + 4800 p4 + 1200 p5)
#define NP3     19200
#define NP4     24000          // cumulative end of p4
#define NB      2              // batch
#define M_GT    40             // gt boxes per image
#define GT_F    (M_GT * 5)     // 200 floats of gt data per image
#define IMG     640.0f
#define SCALE_CLAMP 3.332204510175204f   // log(224/8) = ln(28)

__global__ __launch_bounds__(256)
void rpn_fused_kernel(const float* __restrict__ loc_p3,
                      const float* __restrict__ loc_p4,
                      const float* __restrict__ loc_p5,
                      const float* __restrict__ deltas,     // (B, NANCH, 4)
                      const float* __restrict__ scores,     // (B, NANCH)
                      const float* __restrict__ gt_boxes,   // (B, 40, 5)
                      float* __restrict__ out_boxes,        // (B, NANCH, 4)
                      float* __restrict__ out_probs,        // (B, NANCH)
                      float* __restrict__ out_matched,      // (B, NANCH, 5)
                      float* __restrict__ out_gtd)          // (B, NANCH, 4)
{
    const int b   = blockIdx.y;
    const int tid = threadIdx.x;
    const int i   = blockIdx.x * blockDim.x + tid;

    // ---- stage this image's 40x5 GT table into LDS via CDNA5 async copy ----
    __shared__ float s_gt[GT_F];
    const float* gsrc = gt_boxes + (size_t)b * GT_F;

#if defined(__HIP_DEVICE_COMPILE__) && defined(__gfx1250__)
    if (tid < GT_F) {
        // low 32 bits of a generic pointer to LDS == workgroup-relative LDS offset
        uint32_t lds_addr   = (uint32_t)(uintptr_t)(&s_gt[tid]);
        const float* gaddr  = gsrc + tid;
        asm volatile("global_load_async_to_lds_b32 %0, %1, off"
                     :: "v"(lds_addr), "v"(gaddr)
                     : "memory");
    }
    asm volatile("s_wait_asynccnt 0x0" ::: "memory");   // ASYNCcnt == 0 for this wave
#else
    if (tid < GT_F) s_gt[tid] = gsrc[tid];
#endif
    __syncthreads();

    if (i >= NANCH) return;

    // ---- regenerate this anchor on the fly ----
    float cx, cy, stride;
    int ar;
    if (i < NP3) {
        int l = i / 3;            ar = i - l * 3;
        cx = loc_p3[l * 2 + 0];   cy = loc_p3[l * 2 + 1];  stride = 8.0f;
    } else if (i < NP4) {
        int j = i - NP3; int l = j / 3; ar = j - l * 3;
        cx = loc_p4[l * 2 + 0];   cy = loc_p4[l * 2 + 1];  stride = 16.0f;
    } else {
        int j = i - NP4; int l = j / 3; ar = j - l * 3;
        cx = loc_p5[l * 2 + 0];   cy = loc_p5[l * 2 + 1];  stride = 32.0f;
    }
    const float base = 4.0f * stride;            // STRIDE_SCALE * stride
    const float area = base * base;
    const float arv  = (ar == 0) ? 0.5f : ((ar == 1) ? 1.0f : 2.0f);
    const float ws   = sqrtf(area / arv);
    const float hs   = sqrtf(area * arv);
    const float ax1 = cx - 0.5f * ws, ay1 = cy - 0.5f * hs;
    const float ax2 = cx + 0.5f * ws, ay2 = cy + 0.5f * hs;
    const float acx = (ax1 + ax2) * 0.5f, acy = (ay1 + ay2) * 0.5f;
    const float aw  = ax2 - ax1,          ah  = ay2 - ay1;

    const size_t bi = (size_t)b * NANCH + i;

    // ---- 1) apply_deltas + clip ----
    {
        float4 d = *(const float4*)(deltas + bi * 4);
        float dw = fminf(d.z, SCALE_CLAMP);
        float dh = fminf(d.w, SCALE_CLAMP);
        float pcx = acx + d.x * aw;
        float pcy = acy + d.y * ah;
        float pw  = aw * expf(dw);
        float ph  = ah * expf(dh);
        float bx1 = fminf(fmaxf(pcx - 0.5f * pw, 0.0f), IMG);
        float by1 = fminf(fmaxf(pcy - 0.5f * ph, 0.0f), IMG);
        float bx2 = fminf(fmaxf(pcx + 0.5f * pw, 0.0f), IMG);
        float by2 = fminf(fmaxf(pcy + 0.5f * ph, 0.0f), IMG);
        *(float4*)(out_boxes + bi * 4) = make_float4(bx1, by1, bx2, by2);
    }

    // ---- 2) sigmoid(scores) ----
    {
        float s = scores[bi];
        out_probs[bi] = 1.0f / (1.0f + expf(-s));
    }

    // ---- 3) IoU match vs 40 GT boxes (in LDS) ----
    const float a_area = aw * ah;
    float best = -__builtin_huge_valf();
    int   bestIdx = 0;
    bool  anyValid = false;
    #pragma unroll 8
    for (int j = 0; j < M_GT; ++j) {
        const float g0 = s_gt[j * 5 + 0];
        const float g1 = s_gt[j * 5 + 1];
        const float g2 = s_gt[j * 5 + 2];
        const float g3 = s_gt[j * 5 + 3];
        const float g4 = s_gt[j * 5 + 4];
        const bool valid = (g4 != -1.0f);
        anyValid |= valid;
        float m;
        if (valid) {
            float area2 = (g2 - g0) * (g3 - g1);
            float ltx = fmaxf(ax1, g0), lty = fmaxf(ay1, g1);
            float rbx = fminf(ax2, g2), rby = fminf(ay2, g3);
            float w = fmaxf(rbx - ltx, 0.0f);
            float h = fmaxf(rby - lty, 0.0f);
            float inter = w * h;
            float uni   = a_area + area2 - inter;
            m = inter / (uni + 1e-6f);
        } else {
            m = -1.0f;
        }
        if (m > best) { best = m; bestIdx = j; }   // first-max == jnp.argmax
    }

    float m0, m1, m2, m3, m4;
    if (!anyValid) {
        m0 = m1 = m2 = m3 = m4 = -1.0f;            // "fake" row
    } else if (best <= 0.3f) {
        m0 = m1 = m2 = m3 = m4 = -1.0f;            // background
    } else if (best < 0.6f) {
        m0 = m1 = m2 = m3 = m4 = -100000000.0f;    // neutral
    } else {
        m0 = s_gt[bestIdx * 5 + 0];
        m1 = s_gt[bestIdx * 5 + 1];
        m2 = s_gt[bestIdx * 5 + 2];
        m3 = s_gt[bestIdx * 5 + 3];
        m4 = s_gt[bestIdx * 5 + 4];
    }
    {
        float* om = out_matched + bi * 5;
        om[0] = m0; om[1] = m1; om[2] = m2; om[3] = m3; om[4] = m4;
    }

    // ---- 4) get_deltas_from_anchors(anchors, matched[:, :4]) ----
    {
        float o0, o1, o2, o3;
        const bool bg = (m0 == -1.0f) && (m1 == -1.0f) &&
                        (m2 == -1.0f) && (m3 == -1.0f);
        if (bg) {
            o0 = o1 = o2 = o3 = -100000000.0f;
        } else {
            float gcx = (m0 + m2) * 0.5f;
            float gcy = (m1 + m3) * 0.5f;
            float gw  = m2 - m0;
            float gh  = m3 - m1;
            o0 = (gcx - acx) / aw;
            o1 = (gcy - acy) / ah;
            o2 = logf(fmaxf(gw / aw, 1e-12f));
            o3 = logf(fmaxf(gh / ah, 1e-12f));
        }
        *(float4*)(out_gtd + bi * 4) = make_float4(o0, o1, o2, o3);
    }
}

extern "C" void kernel_launch(void* const* d_in, const int* in_sizes, int n_in,
                              void* d_out, int out_size, void* d_ws, size_t ws_size,
                              hipStream_t stream) {
    const float* loc3   = (const float*)d_in[0];
    const float* loc4   = (const float*)d_in[1];
    const float* loc5   = (const float*)d_in[2];
    const float* deltas = (const float*)d_in[3];
    const float* scores = (const float*)d_in[4];
    const float* gt     = (const float*)d_in[5];

    float* out         = (float*)d_out;
    float* out_boxes   = out;                                   // 2*25200*4
    float* out_probs   = out_boxes + (size_t)NB * NANCH * 4;    // 2*25200
    float* out_matched = out_probs + (size_t)NB * NANCH;        // 2*25200*5
    float* out_gtd     = out_matched + (size_t)NB * NANCH * 5;  // 2*25200*4

    dim3 block(256);
    dim3 grid((NANCH + 255) / 256, NB);
    rpn_fused_kernel<<<grid, block, 0, stream>>>(
        loc3, loc4, loc5, deltas, scores, gt,
        out_boxes, out_probs, out_matched, out_gtd);
}